// KNNConv_55542517072445
// MI455X (gfx1250) — compile-verified
//
#include <hip/hip_runtime.h>
#include <hip/hip_bf16.h>
#include <math.h>

typedef __attribute__((ext_vector_type(2))) float v2f;
typedef __attribute__((ext_vector_type(8))) float v8f;

#define B_     2
#define D_     32
#define N_     65536
#define K_     16
#define OUT_   64
#define C_     544          // (K+1)*D
#define EPS_   1e-5f
#define SLOPE_ 0.01f
#define ROW_   548          // LDS row stride in floats (548 mod 64 = 36 -> conflict-free, 16B aligned)
#define NODES_ 32           // nodes per workgroup (2 WMMA node-subtiles)

// ---------------------------------------------------------------------------
// Kernel 1: transpose x (B, D, N) -> xT (B, N, D) so each gathered column is
// one contiguous 128B line (L2-resident; avoids 67M scattered 4B reads).
// ---------------------------------------------------------------------------
__global__ __launch_bounds__(256)
void transpose_kernel(const float* __restrict__ x, float* __restrict__ xT) {
    __shared__ float tile[32][33];
    const int b  = blockIdx.z;
    const int n0 = blockIdx.x * 32;
    const int tx = threadIdx.x;          // 32
    const int ty = threadIdx.y;          // 8
    const float* xb  = x  + (size_t)b * D_ * N_;
    float*       xTb = xT + (size_t)b * N_ * D_;
    #pragma unroll
    for (int i = 0; i < 32; i += 8)
        tile[ty + i][tx] = xb[(size_t)(ty + i) * N_ + (n0 + tx)];   // [d][n]
    __syncthreads();
    #pragma unroll
    for (int i = 0; i < 32; i += 8)
        xTb[(size_t)(n0 + ty + i) * D_ + tx] = tile[tx][ty + i];    // xT[n][d]
}

// ---------------------------------------------------------------------------
// Kernel 2: fused gather + f32 WMMA GEMM (+bias). One WG = 32 nodes x all 64
// output channels for one batch. 8 waves: wave -> (node subtile, chan tile).
// A (16 nodes x 4 chans) from LDS feat tile, B (4 chans x 16 outs) from W.
// ---------------------------------------------------------------------------
__global__ __launch_bounds__(256)
void knn_gemm_kernel(const float* __restrict__ xT,
                     const int*   __restrict__ idx1,   // edge_index[1]: (B, N, K)
                     const float* __restrict__ W,      // (OUT, 544) row-major
                     const float* __restrict__ bW,
                     float*       __restrict__ out) {  // (B, OUT, N) pre-BN
    extern __shared__ float feat[];                    // NODES_ * ROW_ floats
    const int b        = blockIdx.y;
    const int nodeBase = blockIdx.x * NODES_;
    const int tid      = threadIdx.x;

    // ---- gather phase: build feat[node][kp*32 + d] ----
    const float* xTb  = xT   + (size_t)b * N_ * D_;
    const int*   idxb = idx1 + (size_t)b * N_ * K_;
    for (int p = tid; p < NODES_ * (K_ + 1); p += 256) {
        const int nl = p / (K_ + 1);
        const int kp = p - nl * (K_ + 1);
        const int ng = nodeBase + nl;
        const int src = (kp == 0) ? ng : idxb[(size_t)ng * K_ + (kp - 1)];
        const float4* s   = (const float4*)(xTb + (size_t)src * D_);
        float4*       dst = (float4*)(feat + nl * ROW_ + kp * D_);
        #pragma unroll
        for (int q = 0; q < D_ / 4; ++q) dst[q] = s[q];
    }
    __syncthreads();

    // ---- WMMA phase ----
    const int wave = tid >> 5;
    const int lane = tid & 31;
    const int nt   = wave >> 2;      // node subtile 0..1
    const int ct   = wave & 3;       // channel tile 0..3
    const int hi   = lane >> 4;      // K-half select (A/B layout)
    const int lo   = lane & 15;      // M (A) / N (B) index

    const float* Abase = feat + (nt * 16 + lo) * ROW_ + hi * 2;
    const int    ch    = ct * 16 + lo;
    const float* Bbase = W + (size_t)ch * C_ + hi * 2;
    __builtin_prefetch(Bbase, 0, 1);

    v8f acc = {};
    #pragma unroll 4
    for (int k4 = 0; k4 < C_ / 4; ++k4) {
        v2f a = *(const v2f*)(Abase + 4 * k4);
        v2f w = *(const v2f*)(Bbase + 4 * k4);
        // D = A(16x4 f32) * B(4x16 f32) + C, full fp32 accumulate
        acc = __builtin_amdgcn_wmma_f32_16x16x4_f32(
            /*neg_a=*/false, a, /*neg_b=*/false, w,
            /*c_mod=*/(short)0, acc, /*reuse_a=*/false, /*reuse_b=*/false);
    }

    const float bias = bW[ch];
    // D layout: VGPR r, lane half hi -> node = base + r + 8*hi ; N = lo
    const size_t ob = (size_t)b * OUT_ * N_ + (size_t)ch * N_
                    + (size_t)(nodeBase + nt * 16 + 8 * hi);
    #pragma unroll
    for (int r = 0; r < 8; ++r) out[ob + r] = acc[r] + bias;
}

// ---------------------------------------------------------------------------
// Kernel 3: per-channel BN stats (mean, rstd) with a fixed-order tree
// reduction -> deterministic across replays (no FP atomics).
// ---------------------------------------------------------------------------
__global__ __launch_bounds__(256)
void bn_stats_kernel(const float* __restrict__ out, float* __restrict__ stats) {
    __shared__ float s1s[256], s2s[256];
    const int ch  = blockIdx.x;
    const int tid = threadIdx.x;
    float s1 = 0.f, s2 = 0.f;
    for (int b = 0; b < B_; ++b) {
        const float* row = out + (size_t)b * OUT_ * N_ + (size_t)ch * N_;
        for (int i = tid * 4; i < N_; i += 256 * 4) {
            float4 v = *(const float4*)(row + i);
            s1 += v.x + v.y + v.z + v.w;
            s2 += v.x * v.x + v.y * v.y + v.z * v.z + v.w * v.w;
        }
    }
    s1s[tid] = s1; s2s[tid] = s2;
    __syncthreads();
    for (int off = 128; off > 0; off >>= 1) {
        if (tid < off) { s1s[tid] += s1s[tid + off]; s2s[tid] += s2s[tid + off]; }
        __syncthreads();
    }
    if (tid == 0) {
        const float inv  = 1.0f / (float)(B_ * N_);
        const float mean = s1s[0] * inv;
        const float var  = s2s[0] * inv - mean * mean;
        stats[ch]        = mean;
        stats[OUT_ + ch] = rsqrtf(var + EPS_);
    }
}

// ---------------------------------------------------------------------------
// Kernel 4: in-place normalize + gamma/beta + LeakyReLU (float4 streams).
// ---------------------------------------------------------------------------
__global__ __launch_bounds__(256)
void bn_apply_kernel(float* __restrict__ out, const float* __restrict__ stats,
                     const float* __restrict__ gamma, const float* __restrict__ beta) {
    const size_t i4 = ((size_t)blockIdx.x * 256 + threadIdx.x) * 4;
    const int ch    = (int)((i4 >> 16) & (OUT_ - 1));   // (i4 / N_) % OUT_
    const float mean = stats[ch];
    const float g    = gamma[ch] * stats[OUT_ + ch];
    const float bt   = beta[ch];
    float4 v = *(const float4*)(out + i4);
    float r[4] = {v.x, v.y, v.z, v.w};
    #pragma unroll
    for (int j = 0; j < 4; ++j) {
        float t = (r[j] - mean) * g + bt;
        r[j] = (t >= 0.f) ? t : SLOPE_ * t;
    }
    float4 o; o.x = r[0]; o.y = r[1]; o.z = r[2]; o.w = r[3];
    *(float4*)(out + i4) = o;
}

// ---------------------------------------------------------------------------
extern "C" void kernel_launch(void* const* d_in, const int* in_sizes, int n_in,
                              void* d_out, int out_size, void* d_ws, size_t ws_size,
                              hipStream_t stream) {
    const float* x     = (const float*)d_in[0];   // (B, D, N, 1)
    const int*   eidx  = (const int*)  d_in[1];   // (2, B, N, K)
    const float* W     = (const float*)d_in[2];   // (OUT, 544)
    const float* bW    = (const float*)d_in[3];   // (OUT,)
    const float* gamma = (const float*)d_in[4];
    const float* beta  = (const float*)d_in[5];
    float* out = (float*)d_out;                   // (B, OUT, N, 1)

    // workspace: xT (B*N*D floats = 16 MB) then 128 floats of BN stats
    float* xT    = (float*)d_ws;
    float* stats = xT + (size_t)B_ * D_ * N_;
    const int* idx1 = eidx + (size_t)B_ * N_ * K_;   // edge_index[1]

    transpose_kernel<<<dim3(N_ / 32, 1, B_), dim3(32, 8), 0, stream>>>(x, xT);

    knn_gemm_kernel<<<dim3(N_ / NODES_, B_), 256, NODES_ * ROW_ * sizeof(float),
                      stream>>>(xT, idx1, W, bW, out);

    bn_stats_kernel<<<OUT_, 256, 0, stream>>>(out, stats);

    bn_apply_kernel<<<(B_ * OUT_ * N_) / (4 * 256), 256, 0, stream>>>(
        out, stats, gamma, beta);
}